// FusedMSDeformAttn2d_3539053052121
// MI455X (gfx1250) — compile-verified
//
#include <hip/hip_runtime.h>
#include <hip/hip_bf16.h>

// ---------------------------------------------------------------------------
// Fused MSDeformAttn2d for MI455X (gfx1250, wave32).
//   GEMMs  : V_WMMA_F32_16X16X4_F32 (exact fp32 on the matrix pipe)
//            128x64 block tile, 8 waves, wave tile 16x64 (4 accumulators),
//            LDS A row-major / B transposed, stride-20 padding so every
//            fragment is one conflict-free ds_load_b64 (no repack movs).
//   Sample : wave-per-(n,q,head), lane=channel -> coalesced 128B gathers,
//            val tensor (44.6MB) is L2-resident (192MB L2).
// ---------------------------------------------------------------------------

typedef __attribute__((ext_vector_type(2))) float v2f;
typedef __attribute__((ext_vector_type(8))) float v8f;

#define DIMX    256
#define HEADSX  8
#define LEVELSX 4
#define POINTSX 4
#define DHEAD   32
#define NBATCH  2
#define TOTALT  21760
#define QLEN    21760
#define MTOT    (NBATCH * QLEN)   // 43520 = 340 * 128
#define LDP     20                // padded LDS stride (dwords): even (b64-
                                  // aligned) and 20*m mod 64 distinct for m<16

// -------------------------------- WMMA GEMM --------------------------------
// C[M x ncol] = A[M x 256] * W[256 x ncol] + bias,  fp32 row-major.
// Block: 256 threads = 8 waves; tile 128(M) x 64(N); BK = 16.
// Wave w computes rows [w*16, w*16+16) x all 64 cols (4 WMMA N-tiles).
__global__ __launch_bounds__(256) void gemm_wmma_f32(
    const float* __restrict__ A, const float* __restrict__ W,
    const float* __restrict__ bias, float* __restrict__ C, int ncol)
{
    __shared__ float As[128 * LDP];  // [row m][k], padded      (10 KB)
    __shared__ float Bs[64 * LDP];   // [col n][k], transposed  (5 KB)

    const int t     = threadIdx.x;
    const int lane  = t & 31;
    const int wave  = t >> 5;        // 0..7 = M-tile
    const int m0    = blockIdx.x * 128;
    const int n0    = blockIdx.y * 64;
    const int lhalf = lane >> 4;     // 0: K=k,k+1   1: K=k+2,k+3
    const int l15   = lane & 15;

    v8f acc[4] = {{}, {}, {}, {}};

    // A staging: 128 rows x 16 k as 512 float4s, 2 per thread.
    const int ar0 = t >> 1;                  // wrong granularity; use idx form
    (void)ar0;
    // B staging: 16 k-rows x 64 n-cols, one float4 per thread, transposed.
    const int br = t >> 4;                   // 0..15 (k)
    const int bc = (t & 15) * 4;             // 0..60 (n)

    for (int kb = 0; kb < DIMX; kb += 16) {
        #pragma unroll
        for (int i = 0; i < 2; ++i) {
            const int idx = t + i * 256;     // 0..511
            const int r   = idx >> 2;        // 0..127
            const int c   = (idx & 3) * 4;   // 0,4,8,12
            *(float4*)(As + r * LDP + c) =
                *(const float4*)(A + (size_t)(m0 + r) * DIMX + kb + c);
        }
        const float4 bvec =
            *(const float4*)(W + (size_t)(kb + br) * ncol + n0 + bc);
        Bs[(bc + 0) * LDP + br] = bvec.x;
        Bs[(bc + 1) * LDP + br] = bvec.y;
        Bs[(bc + 2) * LDP + br] = bvec.z;
        Bs[(bc + 3) * LDP + br] = bvec.w;
        __syncthreads();

        #pragma unroll
        for (int kk = 0; kk < 16; kk += 4) {
            const int k0 = kk + lhalf * 2;
            const v2f a = *(const v2f*)(As + (wave * 16 + l15) * LDP + k0);
            #pragma unroll
            for (int nt = 0; nt < 4; ++nt) {
                const v2f b = *(const v2f*)(Bs + (nt * 16 + l15) * LDP + k0);
                acc[nt] = __builtin_amdgcn_wmma_f32_16x16x4_f32(
                    false, a, false, b, (short)0, acc[nt], false, false);
            }
        }
        __syncthreads();
    }

    // C/D layout: VGPR j -> M = j + 8*lhalf (within 16), N = l15
    const int rowb = m0 + wave * 16 + lhalf * 8;
    #pragma unroll
    for (int nt = 0; nt < 4; ++nt) {
        const int col = n0 + nt * 16 + l15;
        const float bv = bias[col];
        #pragma unroll
        for (int j = 0; j < 8; ++j)
            C[(size_t)(rowb + j) * ncol + col] = acc[nt][j] + bv;
    }
}

// --------------------------- sampling + attention ---------------------------
// One wave per (n, q, head); lane = channel (D_HEAD == 32 == wave32).
// Per-lane redundant scalar math; gathers are 128B coalesced from L2-resident
// val. Corner-validity branches are wave-uniform.
__global__ __launch_bounds__(256) void msda_sample(
    const float* __restrict__ p,      // (N,Q,4,2)
    const float* __restrict__ val,    // (N,TOTAL,256) = (N,TOTAL,H,32)
    const float* __restrict__ logits, // (N,Q,H,16)
    const float* __restrict__ off,    // (N,Q,H,4,4,2)
    float* __restrict__ tmp)          // (N,Q,256)
{
    const int LVL_H[4] = {128, 64, 32, 16};
    const int LVL_W[4] = {128, 64, 32, 16};
    const int LVL_S[4] = {0, 16384, 20480, 21504};

    const int wid  = blockIdx.x * 8 + (threadIdx.x >> 5);
    const int lane = threadIdx.x & 31;
    const int h = wid & (HEADSX - 1);
    const int q = (wid >> 3) % QLEN;
    const int n = wid / (HEADSX * QLEN);

    const size_t nq = (size_t)n * QLEN + q;

    // softmax over the 16 (level,point) logits of this head
    const float* lg = logits + (nq * HEADSX + h) * (LEVELSX * POINTSX);
    float w[16];
    float mx = -1e30f;
    #pragma unroll
    for (int i = 0; i < 16; ++i) { w[i] = lg[i]; mx = fmaxf(mx, w[i]); }
    float s = 0.f;
    #pragma unroll
    for (int i = 0; i < 16; ++i) { w[i] = __expf(w[i] - mx); s += w[i]; }
    const float inv = 1.f / s;

    const float* ofs = off + (nq * HEADSX + h) * (LEVELSX * POINTSX * 2);
    const float* pp  = p + nq * (LEVELSX * 2);
    const float* vb  = val + (size_t)n * TOTALT * DIMX + h * DHEAD + lane;

    float acc = 0.f;
    #pragma unroll
    for (int l = 0; l < LEVELSX; ++l) {
        const int   Hh = LVL_H[l], Ww = LVL_W[l], s0 = LVL_S[l];
        const float px = pp[l * 2 + 0];
        const float py = pp[l * 2 + 1];
        #pragma unroll
        for (int pt = 0; pt < POINTSX; ++pt) {
            const float ox = ofs[(l * POINTSX + pt) * 2 + 0];
            const float oy = ofs[(l * POINTSX + pt) * 2 + 1];
            // loc = p + off/[W,H]; x = loc.x*W - 0.5
            const float x = px * (float)Ww + ox - 0.5f;
            const float y = py * (float)Hh + oy - 0.5f;
            const float x0f = floorf(x), y0f = floorf(y);
            const float dx = x - x0f, dy = y - y0f;
            const int x0 = (int)x0f, y0 = (int)y0f;
            const int x1 = x0 + 1,  y1 = y0 + 1;
            const float aw = w[l * POINTSX + pt] * inv;

            const float w00 = aw * (1.f - dx) * (1.f - dy);
            const float w10 = aw * dx * (1.f - dy);
            const float w01 = aw * (1.f - dx) * dy;
            const float w11 = aw * dx * dy;

            if (x0 >= 0 && x0 < Ww && y0 >= 0 && y0 < Hh)
                acc += w00 * vb[(size_t)(s0 + y0 * Ww + x0) * DIMX];
            if (x1 >= 0 && x1 < Ww && y0 >= 0 && y0 < Hh)
                acc += w10 * vb[(size_t)(s0 + y0 * Ww + x1) * DIMX];
            if (x0 >= 0 && x0 < Ww && y1 >= 0 && y1 < Hh)
                acc += w01 * vb[(size_t)(s0 + y1 * Ww + x0) * DIMX];
            if (x1 >= 0 && x1 < Ww && y1 >= 0 && y1 < Hh)
                acc += w11 * vb[(size_t)(s0 + y1 * Ww + x1) * DIMX];
        }
    }
    tmp[nq * DIMX + h * DHEAD + lane] = acc;
}

// ------------------------------- launcher ----------------------------------
extern "C" void kernel_launch(void* const* d_in, const int* in_sizes, int n_in,
                              void* d_out, int out_size, void* d_ws, size_t ws_size,
                              hipStream_t stream) {
    const float* q    = (const float*)d_in[0];
    const float* p    = (const float*)d_in[1];
    const float* v    = (const float*)d_in[2];
    const float* Wv   = (const float*)d_in[3];
    const float* bv   = (const float*)d_in[4];
    const float* Woff = (const float*)d_in[5];
    const float* boff = (const float*)d_in[6];
    const float* Ww   = (const float*)d_in[7];
    const float* bw   = (const float*)d_in[8];
    const float* Wout = (const float*)d_in[9];
    const float* bout = (const float*)d_in[10];
    // d_in[11]=shapes, d_in[12]=level_index: compile-time constants, unused.

    float* out    = (float*)d_out;
    float* ws     = (float*)d_ws;
    float* val    = ws;                                       // MTOT*256
    float* lgts   = val  + (size_t)MTOT * DIMX;               // MTOT*128
    float* offb   = lgts + (size_t)MTOT * 128;                // MTOT*256
    float* tmp    = offb + (size_t)MTOT * DIMX;               // MTOT*256

    dim3 blk(256);
    // val = v @ Wv + bv
    gemm_wmma_f32<<<dim3(MTOT / 128, DIMX / 64), blk, 0, stream>>>(
        v, Wv, bv, val, DIMX);
    // logits = q @ Ww + bw
    gemm_wmma_f32<<<dim3(MTOT / 128, 128 / 64), blk, 0, stream>>>(
        q, Ww, bw, lgts, 128);
    // off = q @ Woff + boff
    gemm_wmma_f32<<<dim3(MTOT / 128, DIMX / 64), blk, 0, stream>>>(
        q, Woff, boff, offb, DIMX);
    // softmax + bilinear sampling + attention accumulate
    msda_sample<<<dim3(MTOT * HEADSX / 8), blk, 0, stream>>>(
        p, val, lgts, offb, tmp);
    // out = tmp @ Wout + bout
    gemm_wmma_f32<<<dim3(MTOT / 128, DIMX / 64), blk, 0, stream>>>(
        tmp, Wout, bout, out, DIMX);
}